// HybridSamplerQNN_65481071403693
// MI455X (gfx1250) — compile-verified
//
#include <hip/hip_runtime.h>
#include <math.h>

typedef __attribute__((ext_vector_type(2))) float v2f;
typedef __attribute__((ext_vector_type(8))) float v8f;

#define BN_EPS 1e-5f
#define NBLOCKS 4096

// ---- branchless fast math ---------------------------------------------------

static __device__ inline float fast_rcp(float x) {
#if __has_builtin(__builtin_amdgcn_rcpf)
  return __builtin_amdgcn_rcpf(x);        // v_rcp_f32
#else
  return 1.0f / x;
#endif
}

// Native V_TANH_F32 if the toolchain exposes it; otherwise a branchless
// exp-based tanh: 1 - 2/(e^{2x}+1). Saturates correctly at +/-inf, no exec
// divergence (unlike ocml tanhf which emits a two-path branch per element).
static __device__ inline float fast_tanh(float x) {
#if __has_builtin(__builtin_amdgcn_tanhf)
  return __builtin_amdgcn_tanhf(x);       // v_tanh_f32
#elif __has_builtin(__builtin_amdgcn_tanh_f32)
  return __builtin_amdgcn_tanh_f32(x);    // v_tanh_f32
#else
  const float e = __expf(2.0f * x);       // v_exp_f32
  return 1.0f - 2.0f * fast_rcp(e + 1.0f);
#endif
}

// ---- WMMA helpers -----------------------------------------------------------

// D = A(16x4 f32) * B(4x16 f32) + C(16x16 f32), wave32 WMMA, full fp32.
static __device__ inline v8f wmma4(v2f a, v2f b, v8f c) {
  return __builtin_amdgcn_wmma_f32_16x16x4_f32(
      /*neg_a=*/false, a, /*neg_b=*/false, b,
      /*c_mod=*/(short)0, c, /*reuse_a=*/false, /*reuse_b=*/false);
}

// A-operand chunk for K = [kbase, kbase+4): lane m (0-15) holds W[m][kbase..kbase+1],
// lane m+16 holds W[m][kbase+2..kbase+3]; zero outside (dout, din).
static __device__ inline v2f load_w_chunk(const float* __restrict__ W, int dout,
                                          int din, int kbase, int lane) {
  int m = lane & 15;
  int k = kbase + ((lane >> 4) << 1);
  v2f a; a.x = 0.0f; a.y = 0.0f;
  if (m < dout) {
    if (k < din)     a.x = W[m * din + k];
    if (k + 1 < din) a.y = W[m * din + k + 1];
  }
  return a;
}

// C-operand = bias broadcast across the 16 batch columns; upper-half (M=8..15) zero.
static __device__ inline v8f load_bias_c(const float* __restrict__ bias, int dout,
                                         int lane) {
  v8f c = {};
  if (lane < 16) {
#pragma unroll
    for (int j = 0; j < 8; ++j) c[j] = (j < dout) ? bias[j] : 0.0f;
  }
  return c;
}

// B-operand chunk for K = [kbase, kbase+4) from activation vector h[8]
// (h[j] = feature j of batch row `lane` for lanes 0-15).
// Upper-half lanes need features kbase+2/kbase+3 of row (lane-16): one bpermute each.
static __device__ inline v2f make_b(const float h[8], int kbase, int lane) {
  float hi0 = __shfl(h[kbase + 2], lane & 15, 32);
  float hi1 = __shfl(h[kbase + 3], lane & 15, 32);
  v2f b;
  b.x = (lane < 16) ? h[kbase + 0] : hi0;
  b.y = (lane < 16) ? h[kbase + 1] : hi1;
  return b;
}

template <int DOUT, int DIN>
static __device__ inline void layer(float h[8], v2f a0, v2f a1, const v8f& c,
                                    int lane) {
  v2f b0 = make_b(h, 0, lane);
  v8f d = wmma4(a0, b0, c);
  if (DIN > 4) {                      // second K-chunk accumulates into D
    v2f b1v = make_b(h, 4, lane);
    d = wmma4(a1, b1v, d);
  }
#pragma unroll
  for (int j = 0; j < 8; ++j) h[j] = (j < DOUT) ? fast_tanh(d[j]) : 0.0f;
}

// ---- kernels ----------------------------------------------------------------

__global__ void __launch_bounds__(256) mlp_wmma_kernel(
    const float* __restrict__ x,
    const float* __restrict__ W1, const float* __restrict__ b1,
    const float* __restrict__ W2, const float* __restrict__ b2,
    const float* __restrict__ W3, const float* __restrict__ b3,
    const float* __restrict__ W4, const float* __restrict__ b4,
    const float* __restrict__ W5, const float* __restrict__ b5,
    const float* __restrict__ W6, const float* __restrict__ b6,
    const float* __restrict__ Wh, const float* __restrict__ bh,
    float* __restrict__ out, float* __restrict__ partials, int numTiles) {
  const int tid  = threadIdx.x;
  const int lane = tid & 31;
  const int wavesPerBlock = blockDim.x >> 5;
  const int waveGlobal = blockIdx.x * wavesPerBlock + (tid >> 5);
  const int waveStride = gridDim.x * wavesPerBlock;

  // Hoisted weight A-operands (reused across all tiles of this wave).
  v2f aW1  = load_w_chunk(W1, 8, 2, 0, lane);
  v2f aW2a = load_w_chunk(W2, 8, 8, 0, lane), aW2b = load_w_chunk(W2, 8, 8, 4, lane);
  v2f aW3a = load_w_chunk(W3, 6, 8, 0, lane), aW3b = load_w_chunk(W3, 6, 8, 4, lane);
  v2f aW4a = load_w_chunk(W4, 4, 6, 0, lane), aW4b = load_w_chunk(W4, 4, 6, 4, lane);
  v2f aW5  = load_w_chunk(W5, 3, 4, 0, lane);
  v2f aW6  = load_w_chunk(W6, 3, 3, 0, lane);
  v2f aWh  = load_w_chunk(Wh, 2, 3, 0, lane);
  v8f c1 = load_bias_c(b1, 8, lane);
  v8f c2 = load_bias_c(b2, 8, lane);
  v8f c3 = load_bias_c(b3, 6, lane);
  v8f c4 = load_bias_c(b4, 4, lane);
  v8f c5 = load_bias_c(b5, 3, lane);
  v8f c6 = load_bias_c(b6, 3, lane);
  v8f ch = load_bias_c(bh, 2, lane);

  float s1 = 0.0f, s2 = 0.0f;

  for (int tile = waveGlobal; tile < numTiles; tile += waveStride) {
    const int row = tile * 16 + (lane & 15);
    float h[8];
#pragma unroll
    for (int j = 0; j < 8; ++j) h[j] = 0.0f;
    if (lane < 16) {
      const float2 xv = reinterpret_cast<const float2*>(x)[row];
      h[0] = xv.x;
      h[1] = xv.y;
    }

    layer<8, 2>(h, aW1,  aW1,  c1, lane);
    layer<8, 8>(h, aW2a, aW2b, c2, lane);
    layer<6, 8>(h, aW3a, aW3b, c3, lane);
    layer<4, 6>(h, aW4a, aW4b, c4, lane);
    layer<3, 4>(h, aW5,  aW5,  c5, lane);
    layer<3, 3>(h, aW6,  aW6,  c6, lane);

    // Head: logits, then 2-way softmax -> p0 (p1 = 1 - p0).
    v2f bhv = make_b(h, 0, lane);
    v8f d = wmma4(aWh, bhv, ch);
    const float l0 = d[0], l1 = d[1];
    const float p0 = fast_rcp(1.0f + __expf(l1 - l0));

    if (lane < 16) {
      out[(size_t)row * 2] = p0;   // stage p0; finalize pass rewrites both slots
      s1 += p0;
      s2 += p0 * p0;
    }
  }

  // Deterministic per-block reduction of (sum p0, sum p0^2).
  __shared__ float r0[256];
  __shared__ float r1[256];
  r0[tid] = s1;
  r1[tid] = s2;
  __syncthreads();
  for (int s = blockDim.x >> 1; s > 0; s >>= 1) {
    if (tid < s) {
      r0[tid] += r0[tid + s];
      r1[tid] += r1[tid + s];
    }
    __syncthreads();
  }
  if (tid == 0) {
    partials[2 * blockIdx.x + 0] = r0[0];
    partials[2 * blockIdx.x + 1] = r1[0];
  }
}

__global__ void __launch_bounds__(256) reduce_partials_kernel(
    const float* __restrict__ partials, int n, float* __restrict__ acc) {
  __shared__ float r0[256];
  __shared__ float r1[256];
  const int tid = threadIdx.x;
  float s1 = 0.0f, s2 = 0.0f;
  for (int i = tid; i < n; i += 256) {
    s1 += partials[2 * i + 0];
    s2 += partials[2 * i + 1];
  }
  r0[tid] = s1;
  r1[tid] = s2;
  __syncthreads();
  for (int s = 128; s > 0; s >>= 1) {
    if (tid < s) {
      r0[tid] += r0[tid + s];
      r1[tid] += r1[tid + s];
    }
    __syncthreads();
  }
  if (tid == 0) {
    acc[0] = r0[0];
    acc[1] = r1[0];
  }
}

__global__ void __launch_bounds__(256) finalize_kernel(
    float* __restrict__ out, const float* __restrict__ acc,
    const float* __restrict__ gamma, const float* __restrict__ beta, int B) {
  const float S1 = acc[0], S2 = acc[1];
  const float invB  = 1.0f / (float)B;
  const float mean0 = S1 * invB;
  const float var   = S2 * invB - mean0 * mean0;  // var1 == var0 since p1 = 1 - p0
  const float inv   = rsqrtf(var + BN_EPS);
  const float g0 = gamma[0], g1 = gamma[1];
  const float be0 = beta[0], be1 = beta[1];

  const int i      = blockIdx.x * blockDim.x + threadIdx.x;
  const int stride = gridDim.x * blockDim.x;
  for (int r = i; r < B; r += stride) {
    const float p0 = out[2 * (size_t)r];
    const float o0 = (p0 - mean0) * inv * g0 + be0;
    const float o1 = (mean0 - p0) * inv * g1 + be1;  // (p1 - mean1) == (mean0 - p0)
    float2 o;
    o.x = o0;
    o.y = o1;
    reinterpret_cast<float2*>(out)[r] = o;
  }
}

extern "C" void kernel_launch(void* const* d_in, const int* in_sizes, int n_in,
                              void* d_out, int out_size, void* d_ws, size_t ws_size,
                              hipStream_t stream) {
  const float* x  = (const float*)d_in[0];
  const float* W1 = (const float*)d_in[1];  const float* b1 = (const float*)d_in[2];
  const float* W2 = (const float*)d_in[3];  const float* b2 = (const float*)d_in[4];
  const float* W3 = (const float*)d_in[5];  const float* b3 = (const float*)d_in[6];
  const float* W4 = (const float*)d_in[7];  const float* b4 = (const float*)d_in[8];
  const float* W5 = (const float*)d_in[9];  const float* b5 = (const float*)d_in[10];
  const float* W6 = (const float*)d_in[11]; const float* b6 = (const float*)d_in[12];
  const float* Wh = (const float*)d_in[13]; const float* bh = (const float*)d_in[14];
  const float* gamma = (const float*)d_in[15];
  const float* beta  = (const float*)d_in[16];
  float* out = (float*)d_out;

  const int B        = in_sizes[0] / 2;
  const int numTiles = B / 16;

  float* ws       = (float*)d_ws;
  float* acc      = ws;       // 2 floats
  float* partials = ws + 2;   // 2 * NBLOCKS floats

  mlp_wmma_kernel<<<NBLOCKS, 256, 0, stream>>>(
      x, W1, b1, W2, b2, W3, b3, W4, b4, W5, b5, W6, b6, Wh, bh,
      out, partials, numTiles);
  reduce_partials_kernel<<<1, 256, 0, stream>>>(partials, NBLOCKS, acc);
  finalize_kernel<<<2048, 256, 0, stream>>>(out, acc, gamma, beta, B);
}